// FusedChunk_Kernel_58153857188278
// MI455X (gfx1250) — compile-verified
//
#include <hip/hip_runtime.h>

typedef __attribute__((ext_vector_type(2))) float v2f;
typedef __attribute__((ext_vector_type(8))) float v8f;

#define D_DIM 128
#define CH    64
#define SEQ   4096
#define NT    (SEQ / CH)
#define NH    16
#define NB    2
#define EG    4          // D split into 4 column groups (parallelism)
#define GW    32         // columns per group

// LDS row strides, padded +4 floats: (4*m + k) % 64 -> conflict-free A loads,
// and stride*4 bytes stays a multiple of 16 for b128 LDS accesses.
#define SQ_LD 132
#define SK_LD 132
#define SV_LD 36
#define SS_LD 68
#define SH_LD 36

#define SQ_OFF 0
#define SK_OFF (SQ_OFF + 64 * SQ_LD)
#define SV_OFF (SK_OFF + 64 * SK_LD)
#define SS_OFF (SV_OFF + 64 * SV_LD)
#define SH_OFF (SS_OFF + 64 * SS_LD)
#define SMEM_FLOATS (SH_OFF + 128 * SH_LD)   // 28160 floats = 112640 bytes

static __device__ __forceinline__ v8f wmma_f32(v2f a, v2f b, v8f c) {
    return __builtin_amdgcn_wmma_f32_16x16x4_f32(
        /*neg_a=*/false, a, /*neg_b=*/false, b,
        /*c_mod=*/(short)0, c, /*reuse_a=*/false, /*reuse_b=*/false);
}

extern "C" __global__ __launch_bounds__(256, 1)
void fla_fused_chunk_fwd(const float* __restrict__ q,
                         const float* __restrict__ k,
                         const float* __restrict__ v,
                         float* __restrict__ o,
                         float* __restrict__ hout,
                         float scale)
{
    extern __shared__ float smem[];
    float* sQ = smem + SQ_OFF;   // 64 x 128 (scaled q)
    float* sK = smem + SK_OFF;   // 64 x 128
    float* sV = smem + SV_OFF;   // 64 x 32  (this group's V columns)
    float* sS = smem + SS_OFF;   // 64 x 64  (masked scores)
    float* sH = smem + SH_OFF;   // 128 x 32 (recurrent state columns)

    const int tid  = threadIdx.x;
    const int wave = tid >> 5;
    const int lane = tid & 31;
    const int m16  = lane & 15;
    const int kh   = lane >> 4;          // half-wave select (K pairs 0/1 vs 2/3)

    const int g  = blockIdx.x & (EG - 1);
    const int bh = blockIdx.x >> 2;
    const int h  = bh & (NH - 1);
    const int b  = bh >> 4;

    const size_t row_stride = (size_t)NH * D_DIM;                 // 2048 floats
    const float* qbase = q + ((size_t)b * SEQ * NH + h) * D_DIM;  // + s*row_stride + d
    const float* kbase = k + ((size_t)b * SEQ * NH + h) * D_DIM;
    const float* vbase = v + ((size_t)b * SEQ * NH + h) * D_DIM;

    // lower-triangular S tile list (r >= c), 10 tiles across 8 waves
    const int SR[10] = {0, 1, 1, 2, 2, 2, 3, 3, 3, 3};
    const int SC[10] = {0, 0, 1, 0, 1, 2, 0, 1, 2, 3};

    // zero recurrent state
    for (int i = tid; i < 128 * SH_LD; i += 256) sH[i] = 0.0f;
    __syncthreads();

    const int r_o  = wave & 3;        // this wave's O row tile
    const int eco  = (wave >> 2) * 16; // this wave's column sub-tile in group

    for (int t = 0; t < NT; ++t) {
        const size_t srow = (size_t)t * CH;

        // ---- stage Q (scaled), K, V into LDS ----
        #pragma unroll
        for (int j = 0; j < 8; ++j) {
            int f   = tid + j * 256;          // 0..2047 float4s
            int row = f >> 5;                 // 32 float4 per 128-wide row
            int c4  = (f & 31) << 2;
            float4 qv = *(const float4*)(qbase + (srow + row) * row_stride + c4);
            qv.x *= scale; qv.y *= scale; qv.z *= scale; qv.w *= scale;
            *(float4*)(sQ + row * SQ_LD + c4) = qv;
            float4 kv = *(const float4*)(kbase + (srow + row) * row_stride + c4);
            *(float4*)(sK + row * SK_LD + c4) = kv;
        }
        #pragma unroll
        for (int j = 0; j < 2; ++j) {
            int f   = tid + j * 256;          // 0..511 float4s
            int row = f >> 3;                 // 8 float4 per 32-wide row
            int c4  = (f & 7) << 2;
            float4 vv = *(const float4*)(vbase + (srow + row) * row_stride + g * GW + c4);
            *(float4*)(sV + row * SV_LD + c4) = vv;
        }
        __syncthreads();

        // ---- S = (q*scale) K^T with causal mask (only r >= c tiles) ----
        #pragma unroll
        for (int ii = 0; ii < 2; ++ii) {
            int idx = wave + ii * 8;
            if (idx < 10) {
                const int r = SR[idx], c = SC[idx];
                v8f acc = {};
                const float* aBase = sQ + (r * 16 + m16) * SQ_LD + kh * 2;
                const float* bBase = sK + (c * 16 + m16) * SK_LD + kh * 2; // K^T in B-layout == K row-major
                #pragma unroll 8
                for (int k0 = 0; k0 < 128; k0 += 4) {
                    v2f a  = *(const v2f*)(aBase + k0);
                    v2f bb = *(const v2f*)(bBase + k0);
                    acc = wmma_f32(a, bb, acc);
                }
                #pragma unroll
                for (int j = 0; j < 8; ++j) {
                    int er = r * 16 + j + 8 * kh;   // row within chunk
                    int ec = c * 16 + m16;          // col within chunk
                    float val = acc[j];
                    if (r == c && er < ec) val = 0.0f;   // tril inclusive
                    sS[er * SS_LD + ec] = val;
                }
            }
        }
        __syncthreads();

        // ---- O tile = S·V (causal-truncated K) + (q*scale)·H ----
        {
            v8f acc = {};
            const float* aS  = sS + (r_o * 16 + m16) * SS_LD + kh * 2;
            const float* aQ  = sQ + (r_o * 16 + m16) * SQ_LD + kh * 2;
            const float* bV0 = sV + (kh * 2 + 0) * SV_LD + eco + m16;
            const float* bV1 = sV + (kh * 2 + 1) * SV_LD + eco + m16;
            const float* bH0 = sH + (kh * 2 + 0) * SH_LD + eco + m16;
            const float* bH1 = sH + (kh * 2 + 1) * SH_LD + eco + m16;

            const int kmax = (r_o + 1) * 16;   // skip zeroed upper-tri tiles
            #pragma unroll 4
            for (int k0 = 0; k0 < kmax; k0 += 4) {
                v2f a = *(const v2f*)(aS + k0);
                v2f bb; bb.x = bV0[k0 * SV_LD]; bb.y = bV1[k0 * SV_LD];
                acc = wmma_f32(a, bb, acc);
            }
            #pragma unroll 8
            for (int k0 = 0; k0 < 128; k0 += 4) {
                v2f a = *(const v2f*)(aQ + k0);
                v2f bb; bb.x = bH0[k0 * SH_LD]; bb.y = bH1[k0 * SH_LD];
                acc = wmma_f32(a, bb, acc);
            }
            #pragma unroll
            for (int j = 0; j < 8; ++j) {
                size_t s_idx = srow + r_o * 16 + j + 8 * kh;
                o[(((size_t)b * SEQ + s_idx) * NH + h) * D_DIM + g * GW + eco + m16] = acc[j];
            }
        }
        __syncthreads();   // O readers of sH done before H writers start

        // ---- H += K^T · V  (wave owns d-row tile = wave, both column sub-tiles) ----
        {
            const int d = wave;   // 0..7
            #pragma unroll
            for (int e2t = 0; e2t < 2; ++e2t) {
                const int ec = e2t * 16;
                float* hTile = sH + ec + m16;
                v8f acc;
                #pragma unroll
                for (int j = 0; j < 8; ++j)
                    acc[j] = hTile[(d * 16 + j + 8 * kh) * SH_LD];
                const float* aK0 = sK + (kh * 2 + 0) * SK_LD + d * 16 + m16; // K^T in A-layout
                const float* aK1 = sK + (kh * 2 + 1) * SK_LD + d * 16 + m16;
                const float* bV0 = sV + (kh * 2 + 0) * SV_LD + ec + m16;
                const float* bV1 = sV + (kh * 2 + 1) * SV_LD + ec + m16;
                #pragma unroll 4
                for (int k0 = 0; k0 < 64; k0 += 4) {
                    v2f a;  a.x  = aK0[k0 * SK_LD]; a.y  = aK1[k0 * SK_LD];
                    v2f bb; bb.x = bV0[k0 * SV_LD]; bb.y = bV1[k0 * SV_LD];
                    acc = wmma_f32(a, bb, acc);
                }
                #pragma unroll
                for (int j = 0; j < 8; ++j)
                    hTile[(d * 16 + j + 8 * kh) * SH_LD] = acc[j];
            }
        }
        __syncthreads();   // protects sQ/sK/sV/sS reuse next chunk
    }

    // ---- write final state hf[b,h,:, g*32 : g*32+32] ----
    #pragma unroll
    for (int j = 0; j < 4; ++j) {
        int f    = tid + j * 256;        // 0..1023 float4s
        int drow = f >> 3;
        int c4   = (f & 7) << 2;
        float4 val = *(const float4*)(sH + drow * SH_LD + c4);
        *(float4*)(hout + (((size_t)(b * NH + h)) * D_DIM + drow) * D_DIM + g * GW + c4) = val;
    }
}

extern "C" void kernel_launch(void* const* d_in, const int* in_sizes, int n_in,
                              void* d_out, int out_size, void* d_ws, size_t ws_size,
                              hipStream_t stream) {
    (void)in_sizes; (void)n_in; (void)out_size; (void)d_ws; (void)ws_size;
    const float* q = (const float*)d_in[0];
    const float* k = (const float*)d_in[1];
    const float* v = (const float*)d_in[2];
    float* o  = (float*)d_out;
    float* hf = o + (size_t)NB * SEQ * NH * D_DIM;   // final state appended after o
    const float scale = 0.088388347648318447f;        // 128^-0.5

    const size_t smem = (size_t)SMEM_FLOATS * sizeof(float);
    hipFuncSetAttribute((const void*)fla_fused_chunk_fwd,
                        hipFuncAttributeMaxDynamicSharedMemorySize, (int)smem);

    dim3 grid(NB * NH * EG);   // 128 workgroups
    dim3 blk(256);             // 8 waves (wave32)
    hipLaunchKernelGGL(fla_fused_chunk_fwd, grid, blk, smem, stream,
                       q, k, v, o, hf, scale);
}